// CPA_learnStep_14027363188938
// MI455X (gfx1250) — compile-verified
//
#include <hip/hip_runtime.h>
#include <math.h>

// CPA_learnStep for MI455X (gfx1250, wave32).
// One workgroup (1024 threads = 32 waves) per batch element.
// Phase A: 16x16x2048 GEMM via V_WMMA_F32_16X16X4_F32, K split across 32 waves.
// Phase B: softmax + 2D cumsum (16x16, in LDS).
// Phase C: 15-step DP; per-thread block[] precomputed in registers (step-invariant),
//          D broadcast from LDS, 4-way partial-max reduction per step.

typedef float v2f __attribute__((ext_vector_type(2)));
typedef float v8f __attribute__((ext_vector_type(8)));

#define TT   16
#define CC   2048
#define NEGC (-1e30f)

__global__ __launch_bounds__(1024)
void cpa_learnstep_kernel(const float* __restrict__ f1,
                          const float* __restrict__ f2,
                          float* __restrict__ out, int B) {
  const int b    = blockIdx.x;
  const int tid  = threadIdx.x;
  const int lane = tid & 31;
  const int wave = tid >> 5;          // 0..31

  __shared__ float s_sc[256];         // scores, later cumsum-j scratch
  __shared__ float s_p[256];          // softmax probs
  __shared__ float s_P[256];          // 2D-cumsum'd pred
  __shared__ float s_D[256];          // DP state D_k
  __shared__ float s_pmax[1024];      // per-sub partial maxes
  __shared__ float s_final[16];       // D_k[15,15] for k=0..15

  if (tid < 256) s_sc[tid] = 0.0f;
  __syncthreads();

  // ---------------- Phase A: scores = f1[b] @ f2[b]^T / sqrt(C) ----------------
  {
    const int row  = lane & 15;           // M for A, N for B
    const int koff = (lane >> 4) << 1;    // lanes 0-15: K=0..1, lanes 16-31: K=2..3
    const float* Ap = f1 + (size_t)b * TT * CC + (size_t)row * CC;
    const float* Bp = f2 + (size_t)b * TT * CC + (size_t)row * CC;

    v8f acc = {0.f, 0.f, 0.f, 0.f, 0.f, 0.f, 0.f, 0.f};
    const int k0 = wave * (CC / 32);      // 64-wide K slice per wave
    #pragma unroll 4
    for (int kk = 0; kk < CC / 32; kk += 4) {
      v2f av = *(const v2f*)(Ap + k0 + kk + koff);
      v2f bv = *(const v2f*)(Bp + k0 + kk + koff);
      // (neg_a, A, neg_b, B, c_mod, C, reuse_a, reuse_b)
      acc = __builtin_amdgcn_wmma_f32_16x16x4_f32(
              false, av, false, bv, (short)0, acc, false, false);
    }
    const float scale = 0.022097086912079608f;   // 1/sqrt(2048)
    const int mhi = (lane >> 4) << 3;            // lanes 16-31 hold M = r+8
    #pragma unroll
    for (int r = 0; r < 8; ++r)
      atomicAdd(&s_sc[(r + mhi) * 16 + (lane & 15)], acc[r] * scale);
  }
  __syncthreads();

  // ---------------- Phase B: softmax + 2D cumsum + D0 ----------------
  if (tid < 256) {                      // softmax over j per row i
    const int i = tid >> 4;
    float mx = -3.4e38f;
    #pragma unroll
    for (int jj = 0; jj < 16; ++jj) mx = fmaxf(mx, s_sc[(i << 4) + jj]);
    float sum = 0.f;
    #pragma unroll
    for (int jj = 0; jj < 16; ++jj) sum += expf(s_sc[(i << 4) + jj] - mx);
    s_p[tid] = expf(s_sc[tid] - mx) / sum;
  }
  __syncthreads();
  if (tid < 256) {                      // cumsum over j
    const int i = tid >> 4, j = tid & 15;
    float s = 0.f;
    for (int jj = 0; jj <= j; ++jj) s += s_p[(i << 4) + jj];
    s_sc[tid] = s;
  }
  __syncthreads();
  if (tid < 256) {                      // cumsum over i; D0 = P / ((i+1)(j+1))
    const int i = tid >> 4, j = tid & 15;
    float s = 0.f;
    for (int ii = 0; ii <= i; ++ii) s += s_sc[(ii << 4) + j];
    s_P[tid] = s;
    const float d0 = s / (float)((i + 1) * (j + 1));
    s_D[tid] = d0;
    if (tid == 255) s_final[0] = d0;
  }
  __syncthreads();

  // ---------------- Phase C: DP recurrence ----------------
  // thread -> (i,j) = tid&255 ; sub = tid>>8 covers a in [4*sub, 4*sub+4), all b.
  const int ij  = tid & 255;
  const int i   = ij >> 4, j = ij & 15;
  const int sub = tid >> 8;

  float u[16];
  #pragma unroll
  for (int bb = 0; bb < 16; ++bb) u[bb] = s_P[ij] - s_P[(i << 4) + bb];

  float blk[64];                        // step-invariant block values, in VGPRs
  #pragma unroll
  for (int a4 = 0; a4 < 4; ++a4) {
    const int a = (sub << 2) + a4;
    const float paj = s_P[(a << 4) + j];
    #pragma unroll
    for (int bb = 0; bb < 16; ++bb) {
      const float num = (a < i && bb < j)
                        ? (u[bb] - paj + s_P[(a << 4) + bb]) : NEGC;
      int prod = (i - a) * (j - bb);
      if (prod < 1) prod = 1;
      blk[(a4 << 4) + bb] = num / sqrtf((float)prod);
    }
  }

  for (int k = 1; k < 16; ++k) {
    float m0 = -3.4e38f, m1 = -3.4e38f, m2 = -3.4e38f, m3 = -3.4e38f;
    #pragma unroll
    for (int bb = 0; bb < 16; ++bb) {
      const int ab = (sub << 6);        // (sub*4)*16
      m0 = fmaxf(m0, blk[bb]      + s_D[ab + bb]);
      m1 = fmaxf(m1, blk[16 + bb] + s_D[ab + 16 + bb]);
      m2 = fmaxf(m2, blk[32 + bb] + s_D[ab + 32 + bb]);
      m3 = fmaxf(m3, blk[48 + bb] + s_D[ab + 48 + bb]);
    }
    s_pmax[(sub << 8) + ij] = fmaxf(fmaxf(m0, m1), fmaxf(m2, m3));
    __syncthreads();
    if (sub == 0) {
      const float mm = fmaxf(fmaxf(s_pmax[ij], s_pmax[256 + ij]),
                             fmaxf(s_pmax[512 + ij], s_pmax[768 + ij]));
      s_D[ij] = mm;
      if (ij == 255) s_final[k] = mm;
    }
    __syncthreads();
  }

  // ---------------- outputs ----------------
  if (tid == 0) {
    float best = s_final[0]; int arg = 0;
    #pragma unroll
    for (int k = 1; k < 16; ++k)
      if (s_final[k] > best) { best = s_final[k]; arg = k; }
    out[b]     = -best;            // loss_step
    out[B + b] = (float)arg;       // step_num
  }
}

extern "C" void kernel_launch(void* const* d_in, const int* in_sizes, int n_in,
                              void* d_out, int out_size, void* d_ws, size_t ws_size,
                              hipStream_t stream) {
  const float* f1 = (const float*)d_in[0];
  const float* f2 = (const float*)d_in[1];
  float* out = (float*)d_out;
  const int B = in_sizes[0] / (TT * CC);   // 256
  cpa_learnstep_kernel<<<dim3(B), dim3(1024), 0, stream>>>(f1, f2, out, B);
}